// GAT_86474871538315
// MI455X (gfx1250) — compile-verified
//
#include <hip/hip_runtime.h>
#include <math.h>

// ---------------------------------------------------------------------------
// GAT (3-layer) for MI455X / gfx1250.
// GEMMs use V_WMMA_F32_16X16X4_F32 (exact f32 math, matches reference).
// Edge softmax + aggregation are streaming passes with f32/L2 atomics.
// ---------------------------------------------------------------------------

#define NEG_SLOPE 0.2f

typedef float v2f __attribute__((ext_vector_type(2)));
typedef float v8f __attribute__((ext_vector_type(8)));

// ------------------------- order-preserving float<->uint for atomicMax -----
__device__ __forceinline__ unsigned fenc(float f) {
    unsigned u = __float_as_uint(f);
    return (u & 0x80000000u) ? ~u : (u | 0x80000000u);
}
__device__ __forceinline__ float fdec(unsigned k) {
    return __uint_as_float((k & 0x80000000u) ? (k & 0x7FFFFFFFu) : ~k);
}
// encoded(-inf) = ~0xFF800000 = 0x007FFFFF
#define ENC_NEG_INF 0x007FFFFFu

// ------------------------------------------------------------ fill scratch -
__global__ void fill_u32_kernel(unsigned* __restrict__ p, unsigned v, long long n) {
    long long i = (long long)blockIdx.x * blockDim.x + threadIdx.x;
    if (i < n) p[i] = v;
}

// ----------------------------------------------------- WMMA f32 GEMM  -----
// Y[M x NC] = X[M x K] * W[K x NC], all row-major f32. M % 16 == 0, K % 4 == 0.
// blockDim = 256 (8 waves); each wave computes one 16x16 tile.
//
// Out-of-range columns (layer 3, NC=40): B loads are ADDRESS-CLAMPED instead
// of predicated. In the WMMA B layout, column n of B lives only in lanes with
// (lane&15)==n and contributes only to column n of D, which those same lanes
// own and never store when out of range — so clamped (garbage) B values are
// harmless and the inner loop stays branch-free.
__global__ __launch_bounds__(256)
void gemm16x16_wmma_f32(const float* __restrict__ X,
                        const float* __restrict__ W,
                        float* __restrict__ Y,
                        int K, int NC) {
    const int lane = threadIdx.x & 31;
    const int wave = threadIdx.x >> 5;
    const int row0 = blockIdx.x << 4;
    const int col0 = ((blockIdx.y << 3) + wave) << 4;
    if (col0 >= NC) return;                 // wave-uniform exit

    const int m  = lane & 15;               // A row within tile
    const int kh = (lane >> 4) << 1;        // K sub-offset: 0 (lanes 0-15) / 2 (16-31)
    const int n  = lane & 15;               // B/D column within tile
    const int ncol = min(col0 + n, NC - 1); // clamped load column

    const float* xp = X + (size_t)(row0 + m) * K + kh;
    const float* wp = W + (size_t)kh * NC + ncol;

    v8f acc = {0.f, 0.f, 0.f, 0.f, 0.f, 0.f, 0.f, 0.f};
    #pragma unroll 8
    for (int k = 0; k < K; k += 4) {
        v2f a = *(const v2f*)xp;            // X[m][k+kh], X[m][k+kh+1]
        v2f b;
        b.x = wp[0];                        // W[k+kh  ][ncol]
        b.y = wp[NC];                       // W[k+kh+1][ncol]
        acc = __builtin_amdgcn_wmma_f32_16x16x4_f32(
            /*neg_a=*/false, a, /*neg_b=*/false, b,
            /*c_mod=*/(short)0, acc, /*reuse_a=*/false, /*reuse_b=*/false);
        xp += 4;
        wp += (size_t)4 * NC;
    }
    if ((col0 + n) < NC) {
        // D layout: VGPR r -> M = r + (lane>=16 ? 8 : 0), N = lane&15
        const int mbase = row0 + ((lane >> 4) << 3);
        float* yp = Y + (size_t)mbase * NC + col0 + n;
        #pragma unroll
        for (int r = 0; r < 8; ++r) yp[(size_t)r * NC] = acc[r];
    }
}

// -------------------------------------------- per-node attention logits ----
// a_s[n,h] = sum_c xw[n,h,c] * att_src[h,c];  a_d likewise.
__global__ void node_logits_kernel(const float* __restrict__ xw,
                                   const float* __restrict__ att_src,
                                   const float* __restrict__ att_dst,
                                   float* __restrict__ a_s,
                                   float* __restrict__ a_d,
                                   int Nn, int H, int C) {
    int i = blockIdx.x * blockDim.x + threadIdx.x;
    if (i >= Nn * H) return;
    int node = i / H, h = i - node * H;
    const float* xp = xw + (size_t)node * (H * C) + h * C;
    const float* as = att_src + h * C;
    const float* ad = att_dst + h * C;
    float s0 = 0.f, s1 = 0.f;
    for (int c = 0; c < C; ++c) {
        float v = xp[c];
        s0 += v * as[c];
        s1 += v * ad[c];
    }
    a_s[i] = s0;
    a_d[i] = s1;
}

// ------------------------------------------------ edge pass 1: logit+max --
__global__ void edge_logit_max_kernel(const int* __restrict__ src,
                                      const int* __restrict__ dst,
                                      int E0, int Etot,
                                      const float* __restrict__ a_s,
                                      const float* __restrict__ a_d,
                                      float* __restrict__ ebuf,
                                      unsigned* __restrict__ emax,
                                      int logH) {
    unsigned idx = blockIdx.x * blockDim.x + threadIdx.x;
    const int H = 1 << logH;
    if (idx >= ((unsigned)Etot << logH)) return;
    int e = idx >> logH, h = idx & (H - 1);
    int s = (e < E0) ? src[e] : (e - E0);
    int d = (e < E0) ? dst[e] : (e - E0);
    float v = a_s[s * H + h] + a_d[d * H + h];
    v = (v > 0.f) ? v : NEG_SLOPE * v;      // leaky_relu
    ebuf[idx] = v;
    atomicMax(&emax[d * H + h], fenc(v));
}

// ------------------------------------------------ edge pass 2: exp+sum ----
__global__ void edge_exp_kernel(const int* __restrict__ dst,
                                int E0, int Etot,
                                const unsigned* __restrict__ emax,
                                float* __restrict__ ebuf,
                                float* __restrict__ denom,
                                int logH) {
    unsigned idx = blockIdx.x * blockDim.x + threadIdx.x;
    const int H = 1 << logH;
    if (idx >= ((unsigned)Etot << logH)) return;
    int e = idx >> logH, h = idx & (H - 1);
    int d = (e < E0) ? dst[e] : (e - E0);
    float ee = expf(ebuf[idx] - fdec(emax[d * H + h]));
    ebuf[idx] = ee;
    atomicAdd(&denom[d * H + h], ee);
}

// ------------------------------------------------ edge pass 3: normalize --
__global__ void edge_alpha_kernel(const int* __restrict__ dst,
                                  int E0, int Etot,
                                  const float* __restrict__ denom,
                                  float* __restrict__ ebuf,
                                  int logH) {
    unsigned idx = blockIdx.x * blockDim.x + threadIdx.x;
    const int H = 1 << logH;
    if (idx >= ((unsigned)Etot << logH)) return;
    int e = idx >> logH, h = idx & (H - 1);
    int d = (e < E0) ? dst[e] : (e - E0);
    ebuf[idx] = ebuf[idx] / (denom[d * H + h] + 1e-16f);
}

// --------------------------------- edge pass 4: gather * alpha -> scatter --
// One thread handles 4 consecutive channels (float4 gather + 4 atomic adds).
// All index math in 32-bit: max total = Etot * HC4 = 850000*64 < 2^31.
__global__ void edge_agg_kernel(const int* __restrict__ src,
                                const int* __restrict__ dst,
                                int E0, int Etot,
                                const float* __restrict__ alpha,
                                const float* __restrict__ xw,
                                float* __restrict__ agg,
                                int H, int C) {
    const unsigned HC  = (unsigned)(H * C);
    const unsigned HC4 = HC >> 2;
    unsigned idx = blockIdx.x * blockDim.x + threadIdx.x;
    if (idx >= (unsigned)Etot * HC4) return;
    unsigned e  = idx / HC4;
    unsigned q  = idx - e * HC4;
    unsigned c4 = q << 2;
    unsigned h  = c4 / (unsigned)C;
    int s = ((int)e < E0) ? src[e] : ((int)e - E0);
    int d = ((int)e < E0) ? dst[e] : ((int)e - E0);
    float a = alpha[e * (unsigned)H + h];
    const float4 xv = *(const float4*)(xw + (size_t)s * HC + c4);
    float* ap = agg + (size_t)d * HC + c4;
    atomicAdd(ap + 0, a * xv.x);
    atomicAdd(ap + 1, a * xv.y);
    atomicAdd(ap + 2, a * xv.z);
    atomicAdd(ap + 3, a * xv.w);
}

// --------------------------------------------------- epilogue: bias+ELU ---
// HC is a power of two (256) on this path.
__global__ void bias_elu_kernel(const float* __restrict__ agg,
                                const float* __restrict__ bias,
                                float* __restrict__ out,
                                long long total, int HC) {
    long long i = (long long)blockIdx.x * blockDim.x + threadIdx.x;
    if (i >= total) return;
    int c = (int)i & (HC - 1);
    float v = agg[i] + bias[c];
    out[i] = (v > 0.f) ? v : (expf(v) - 1.f);
}

// --------------------------------------- epilogue: bias + log_softmax -----
__global__ void bias_logsoftmax_kernel(const float* __restrict__ agg,
                                       const float* __restrict__ bias,
                                       float* __restrict__ out,
                                       int Nn, int C) {
    int n = blockIdx.x * blockDim.x + threadIdx.x;
    if (n >= Nn) return;
    const float* ap = agg + (size_t)n * C;
    float mx = -INFINITY;
    for (int c = 0; c < C; ++c) {
        float v = ap[c] + bias[c];
        mx = fmaxf(mx, v);
    }
    float sum = 0.f;
    for (int c = 0; c < C; ++c) sum += expf(ap[c] + bias[c] - mx);
    float lse = mx + logf(sum);
    float* op = out + (size_t)n * C;
    for (int c = 0; c < C; ++c) op[c] = ap[c] + bias[c] - lse;
}

// ---------------------------------------------------------------------------
// Host-side: one GAT layer = gemm -> logits -> init -> 4 edge passes -> epi.
// ---------------------------------------------------------------------------
static void run_layer(const float* Xin, const float* Wm,
                      const float* att_s, const float* att_d, const float* bias,
                      float* xw, float* agg, unsigned* emax, float* denom,
                      float* a_sv, float* a_dv, float* ebuf,
                      const int* src, const int* dst, int E0, int Etot,
                      int Nn, int K, int H, int C, int logH,
                      float* outbuf, bool final_layer, hipStream_t stream) {
    const int HC = H * C;
    // 1) xw = Xin @ Wm
    {
        int colTiles = (HC + 15) / 16;
        dim3 grid(Nn / 16, (colTiles + 7) / 8);
        gemm16x16_wmma_f32<<<grid, 256, 0, stream>>>(Xin, Wm, xw, K, HC);
    }
    // 2) per-node logits
    {
        int nh = Nn * H;
        node_logits_kernel<<<(nh + 255) / 256, 256, 0, stream>>>(
            xw, att_s, att_d, a_sv, a_dv, Nn, H, C);
    }
    // 3) init emax / denom / agg
    {
        long long nh = (long long)Nn * H;
        fill_u32_kernel<<<(int)((nh + 255) / 256), 256, 0, stream>>>(emax, ENC_NEG_INF, nh);
        fill_u32_kernel<<<(int)((nh + 255) / 256), 256, 0, stream>>>((unsigned*)denom, 0u, nh);
        long long nhc = (long long)Nn * HC;
        fill_u32_kernel<<<(int)((nhc + 255) / 256), 256, 0, stream>>>((unsigned*)agg, 0u, nhc);
    }
    // 4) edge passes
    {
        int eh = Etot << logH;
        int blocks = (eh + 255) / 256;
        edge_logit_max_kernel<<<blocks, 256, 0, stream>>>(src, dst, E0, Etot,
                                                          a_sv, a_dv, ebuf, emax, logH);
        edge_exp_kernel<<<blocks, 256, 0, stream>>>(dst, E0, Etot, emax, ebuf, denom, logH);
        edge_alpha_kernel<<<blocks, 256, 0, stream>>>(dst, E0, Etot, denom, ebuf, logH);
        long long tq = (long long)Etot * (HC >> 2);
        edge_agg_kernel<<<(int)((tq + 255) / 256), 256, 0, stream>>>(
            src, dst, E0, Etot, ebuf, xw, agg, H, C);
    }
    // 5) epilogue
    if (!final_layer) {
        long long nhc = (long long)Nn * HC;
        bias_elu_kernel<<<(int)((nhc + 255) / 256), 256, 0, stream>>>(agg, bias, outbuf, nhc, HC);
    } else {
        bias_logsoftmax_kernel<<<(Nn + 255) / 256, 256, 0, stream>>>(agg, bias, outbuf, Nn, HC);
    }
}

extern "C" void kernel_launch(void* const* d_in, const int* in_sizes, int n_in,
                              void* d_out, int out_size, void* d_ws, size_t ws_size,
                              hipStream_t stream) {
    const float* x     = (const float*)d_in[0];
    const int*   eidx  = (const int*)d_in[1];
    const float* W1    = (const float*)d_in[2];
    const float* a1s   = (const float*)d_in[3];
    const float* a1d   = (const float*)d_in[4];
    const float* b1    = (const float*)d_in[5];
    const float* W2    = (const float*)d_in[6];
    const float* a2s   = (const float*)d_in[7];
    const float* a2d   = (const float*)d_in[8];
    const float* b2    = (const float*)d_in[9];
    const float* W3    = (const float*)d_in[10];
    const float* a3s   = (const float*)d_in[11];
    const float* a3d   = (const float*)d_in[12];
    const float* b3    = (const float*)d_in[13];

    const int N    = in_sizes[0] / 128;     // F_IN = 128
    const int E0   = in_sizes[1] / 2;       // edge_index is [2, E]
    const int Etot = E0 + N;                // + self loops
    const int* src = eidx;
    const int* dst = eidx + E0;

    // ------------------- workspace layout (all f32 unless noted) ----------
    float* ws = (float*)d_ws;
    const size_t NHC = (size_t)N * 256;
    float*    xw    = ws;                       // [N,256] GEMM output
    float*    agg   = ws + NHC;                 // [N,256] aggregation accumulator
    float*    hbuf  = ws + 2 * NHC;             // [N,256] inter-layer activations
    float*    a_sv  = ws + 3 * NHC;             // [N,4]
    float*    a_dv  = a_sv + (size_t)N * 4;     // [N,4]
    unsigned* emax  = (unsigned*)(a_dv + (size_t)N * 4);  // [N,4] encoded f32
    float*    denom = (float*)(emax + (size_t)N * 4);     // [N,4]
    float*    ebuf  = denom + (size_t)N * 4;    // [Etot,4] edge logits/alpha

    // layer 1: 128 -> 4x64, ELU
    run_layer(x, W1, a1s, a1d, b1, xw, agg, emax, denom, a_sv, a_dv, ebuf,
              src, dst, E0, Etot, N, 128, 4, 64, 2, hbuf, false, stream);
    // layer 2: 256 -> 4x64, ELU (hbuf -> hbuf is safe: gemm consumes it first)
    run_layer(hbuf, W2, a2s, a2d, b2, xw, agg, emax, denom, a_sv, a_dv, ebuf,
              src, dst, E0, Etot, N, 256, 4, 64, 2, hbuf, false, stream);
    // layer 3: 256 -> 1x40, log_softmax -> d_out
    run_layer(hbuf, W3, a3s, a3d, b3, xw, agg, emax, denom, a_sv, a_dv, ebuf,
              src, dst, E0, Etot, N, 256, 1, 40, 0, (float*)d_out, true, stream);
}